// chamferDist_60593398612307
// MI455X (gfx1250) — compile-verified
//
#include <hip/hip_runtime.h>

// Chamfer distance on MI455X (gfx1250, wave32).
// Squared-distance tile via ONE V_WMMA_F32_16X16X4_F32:
//   A row m = (-2qx, -2qy, -2qz, 1)        (built once per wave, NT tiles)
//   B col n = ( rx,   ry,   rz, |r|^2)     (rebuilt per 16-ref tile)
//   C      = 0 (inline-zero, loop invariant)
//   => D[m][n] = -2 q.r + |r|^2 ;  |q|^2 added after the row-min (constant/row).
// Row-min over the ref stream -> nearest-neighbor distance. Launched twice with
// roles swapped (dist1/dist2). Ref dimension split into chunks for occupancy;
// per-chunk partial mins in d_ws, combined by a small reduction kernel.

typedef __attribute__((ext_vector_type(2))) float v2f;
typedef __attribute__((ext_vector_type(8))) float v8f;

#define TILE 16
#define NT   4            // 16-row query tiles per wave -> 64 rows/wave
#define BIGF 3.0e38f

// Raw min ops: inputs are finite, skip fminf's canonicalize (v_max x,x).
__device__ __forceinline__ float min2f(float a, float b) {
    float r;
    asm("v_min_num_f32 %0, %1, %2" : "=v"(r) : "v"(a), "v"(b));
    return r;
}
__device__ __forceinline__ float min3f(float a, float b, float c) {
    float r;
    asm("v_min3_num_f32 %0, %1, %2, %3" : "=v"(r) : "v"(a), "v"(b), "v"(c));
    return r;
}

__global__ __launch_bounds__(256)
void chamfer_nn_kernel(const float* __restrict__ query,
                       const float* __restrict__ ref,
                       float* __restrict__ partial,   // [nChunks][B][nQuery]
                       int nQuery, int nRef, int strips, int B,
                       int tilesPerChunk)
{
    const int tid  = blockIdx.x * blockDim.x + threadIdx.x;
    // Force wave-uniform scalars into SGPRs: divergence analysis can't prove
    // tid>>5 is uniform, which otherwise turns the sweep loop into a
    // divergent (exec-masked, VGPR-controlled) loop.
    const int wave = __builtin_amdgcn_readfirstlane(tid >> 5);
    const int lane = tid & 31;
    const int half = lane >> 4;   // 0: K=0,1 slice   1: K=2,3 slice
    const int l15  = lane & 15;   // M (A) / N (B) within tile

    const int perChunk = B * strips;
    const int ch  = wave / perChunk;
    const int rem = wave % perChunk;
    const int b   = rem / strips;
    const int s   = rem % strips;
    const int row0 = s * (TILE * NT);

    const float* qB = query   + (size_t)b * nQuery * 3;
    const float* rB = ref     + (size_t)b * nRef   * 3;
    float*       oB = partial + ((size_t)ch * B + b) * nQuery;

    // ---- A tiles (once per wave): lane l15 owns query row row0+t*16+l15 ----
    // f32 A 16x4 layout: VGPR0 = {K0 | K2}, VGPR1 = {K1 | K3} across halves.
    v2f   a[NT];
    float sq[NT];
#pragma unroll
    for (int t = 0; t < NT; ++t) {
        const float* qp = qB + (size_t)(row0 + t * TILE + l15) * 3;
        const float qx = qp[0], qy = qp[1], qz = qp[2];
        sq[t] = qx * qx + qy * qy + qz * qz;
        a[t]  = half ? (v2f){-2.0f * qz, 1.0f}
                     : (v2f){-2.0f * qx, -2.0f * qy};
    }

    v8f acc[NT];
#pragma unroll
    for (int t = 0; t < NT; ++t)
        acc[t] = (v8f){BIGF, BIGF, BIGF, BIGF, BIGF, BIGF, BIGF, BIGF};

    const v8f cz = (v8f){0.f, 0.f, 0.f, 0.f, 0.f, 0.f, 0.f, 0.f}; // hoisted

    // ---- Sweep this chunk of reference tiles, 2 tiles per iteration -------
    const int jBeg = ch * tilesPerChunk;           // SGPR-uniform bounds
    const int jEnd = jBeg + tilesPerChunk;
    for (int j = jBeg; j < jEnd; j += 2) {
        const int pj = (j + 16 < jEnd) ? (j + 16) : j;
        __builtin_prefetch(rB + (size_t)(pj * TILE + l15) * 3, 0, 1);

        const float* rp0 = rB + (size_t)((j    ) * TILE + l15) * 3;
        const float* rp1 = rB + (size_t)((j + 1) * TILE + l15) * 3;
        const float r0x = rp0[0], r0y = rp0[1], r0z = rp0[2];
        const float r1x = rp1[0], r1y = rp1[1], r1z = rp1[2];
        const float sr0 = r0x * r0x + r0y * r0y + r0z * r0z;
        const float sr1 = r1x * r1x + r1y * r1y + r1z * r1z;

        // B 4x16: lane = column N; halves carry (K0,K1)/(K2,K3); K3 = |r|^2.
        const v2f b0 = half ? (v2f){r0z, sr0} : (v2f){r0x, r0y};
        const v2f b1 = half ? (v2f){r1z, sr1} : (v2f){r1x, r1y};

#pragma unroll
        for (int t = 0; t < NT; ++t) {
            v8f d0 = __builtin_amdgcn_wmma_f32_16x16x4_f32(
                false, a[t], false, b0, (short)0, cz, false, false);
            v8f d1 = __builtin_amdgcn_wmma_f32_16x16x4_f32(
                false, a[t], false, b1, (short)0, cz, false, false);
#pragma unroll
            for (int k = 0; k < 8; ++k)
                acc[t][k] = min3f(acc[t][k], d0[k], d1[k]);
        }
    }

    // ---- Per-row min across the 16 columns in each half-wave --------------
    // Row v: lanes 0..15 of VGPR v;  row v+8: lanes 16..31 of VGPR v.
#pragma unroll
    for (int t = 0; t < NT; ++t) {
#pragma unroll
        for (int v = 0; v < 8; ++v) {
            float r = acc[t][v];
            r = min2f(r, __shfl_xor(r, 1, 32));
            r = min2f(r, __shfl_xor(r, 2, 32));
            r = min2f(r, __shfl_xor(r, 4, 32));
            r = min2f(r, __shfl_xor(r, 8, 32));
            // add back |q|^2 of the row this half-wave owns (held in lane v+8*half)
            const float sqr = __shfl(sq[t], v + 8 * half, 32);
            if (l15 == 0)
                oB[row0 + t * TILE + v + half * 8] = r + sqr;
        }
    }
}

__global__ __launch_bounds__(256)
void reduce_min_kernel(const float* __restrict__ partial,  // [nChunks][n]
                       float* __restrict__ out, int n, int nChunks)
{
    const int i = blockIdx.x * blockDim.x + threadIdx.x;
    if (i < n) {
        float r = partial[i];
        for (int c = 1; c < nChunks; ++c)
            r = min2f(r, partial[(size_t)c * n + i]);
        out[i] = r;
    }
}

static void run_pass(const float* query, const float* ref, float* out,
                     int nQuery, int nRef, int B, float* ws, int nChunks,
                     hipStream_t stream)
{
    const int strips = nQuery / (TILE * NT);
    const int tilesPerChunk = (nRef / TILE) / nChunks;
    const int waves  = B * strips * nChunks;
    const dim3 block(256);
    float* partial = (nChunks > 1) ? ws : out;

    chamfer_nn_kernel<<<(waves * 32) / 256, block, 0, stream>>>(
        query, ref, partial, nQuery, nRef, strips, B, tilesPerChunk);

    if (nChunks > 1) {
        const int n = B * nQuery;
        reduce_min_kernel<<<(n + 255) / 256, block, 0, stream>>>(
            partial, out, n, nChunks);
    }
}

extern "C" void kernel_launch(void* const* d_in, const int* in_sizes, int n_in,
                              void* d_out, int out_size, void* d_ws, size_t ws_size,
                              hipStream_t stream) {
    (void)n_in; (void)out_size;
    const int B = 4;
    const int N = in_sizes[0] / (B * 3);   // 8192
    const int M = in_sizes[1] / (B * 3);   // 8192

    const float* xyz1 = (const float*)d_in[0];
    const float* xyz2 = (const float*)d_in[1];
    float* out = (float*)d_out;

    const int maxQ = (N > M) ? N : M;
    int nChunks = 4;
    if (ws_size < (size_t)nChunks * B * maxQ * sizeof(float)) nChunks = 1;

    // dist1[b,n] = min_m d(xyz1, xyz2); dist2[b,m] = min_n d(xyz2, xyz1)
    run_pass(xyz1, xyz2, out,                N, M, B, (float*)d_ws, nChunks, stream);
    run_pass(xyz2, xyz1, out + (size_t)B * N, M, N, B, (float*)d_ws, nChunks, stream);
}